// LinearScanModule_5420248727655
// MI455X (gfx1250) — compile-verified
//
#include <hip/hip_runtime.h>

// Linear scan x_t = a_t * x_{t-1} + u_t  over [B=8, T=4096, D=1024], fp32.
// Single-pass chunked scan with deterministic aggregate-only lookback.
// Traffic = read a + read u + write x = 384 MB (the minimum) -> ~16.5us @ 23.3 TB/s.

#define B_ 8
#define T_ 4096
#define D_ 1024
#define TILE_D 64                  // channels per block
#define CHUNK_T 64                 // timesteps per block
#define N_TILES (D_ / TILE_D)      // 16
#define N_CHUNKS (T_ / CHUNK_T)    // 64
#define TPB 256
#define ROWS (TPB / TILE_D)        // 4 t-rows per block
#define TPT (CHUNK_T / ROWS)       // 16 timesteps per thread

// ---- CDNA5 async global->LDS staging (guarded; falls back to plain loads) ----
#if defined(__gfx1250__) && __has_builtin(__builtin_amdgcn_global_load_async_to_lds_b128) && \
    __has_builtin(__builtin_amdgcn_s_wait_asynccnt)
#define USE_ASYNC_LDS 1
#else
#define USE_ASYNC_LDS 0
#endif

typedef int v4i __attribute__((vector_size(16)));

__global__ void zero_flags_kernel(int* __restrict__ flags, int n) {
  int i = blockIdx.x * blockDim.x + threadIdx.x;
  if (i < n) flags[i] = 0;
}

// Fallback if workspace is too small: 1 thread per (b,d) channel, sequential over T.
__global__ __launch_bounds__(256) void scan_seq_kernel(const float* __restrict__ x0,
                                                       const float* __restrict__ a,
                                                       const float* __restrict__ u,
                                                       float* __restrict__ out) {
  int i = blockIdx.x * blockDim.x + threadIdx.x;  // i = b*D + d
  if (i >= B_ * D_) return;
  int b = i >> 10;
  int d = i & (D_ - 1);
  float x = x0[i];
  size_t idx = (size_t)b * T_ * D_ + d;
  for (int t = 0; t < T_; ++t, idx += D_) {
    x = fmaf(a[idx], x, u[idx]);
    out[idx] = x;
  }
}

__global__ __launch_bounds__(TPB) void scan_chunked_kernel(
    const float* __restrict__ x0, const float* __restrict__ a, const float* __restrict__ u,
    float* __restrict__ out, float* __restrict__ aggP, float* __restrict__ aggH,
    int* __restrict__ flags) {
  const int c = blockIdx.x;     // chunk along T (x-fastest => predecessors dispatch first)
  const int tile = blockIdx.y;  // d-tile
  const int b = blockIdx.z;     // batch
  const int tid = threadIdx.x;
  const int dl = tid & (TILE_D - 1);  // channel within tile
  const int row = tid >> 6;           // t-row within chunk (0..3)
  const int d0 = tile * TILE_D;
  const size_t rowStart = ((size_t)b * T_ + (size_t)c * CHUNK_T) * D_ + d0;

  __shared__ float a_s[CHUNK_T * TILE_D];  // [t_local][d_local], 16 KB
  __shared__ float u_s[CHUNK_T * TILE_D];  // 16 KB
  __shared__ float sP[ROWS * TILE_D];
  __shared__ float sH[ROWS * TILE_D];
  __shared__ float sHin[TILE_D];
  __shared__ int sFlag;

  // ---- Stage the 64x64 a/u tiles into LDS ----
#if USE_ASYNC_LDS
  {
    // Each thread async-copies 4x 16B segments per array; 256 threads cover the
    // full 64(t) x 64(d) tile. Generic pointers implicitly addrspacecast to the
    // builtin's global/LDS parameter address spaces.
    const int k = tid >> 2;        // t_local row 0..63 (4 threads per row)
    const int seg0 = tid & 3;      // 16-byte segment within row
    const size_t gRow = rowStart + (size_t)k * D_;
#pragma unroll
    for (int it = 0; it < 4; ++it) {
      const int seg = seg0 + it * 4;  // 0..15, each 4 floats
      __builtin_amdgcn_global_load_async_to_lds_b128(
          (v4i*)(a + gRow + seg * 4), (v4i*)(&a_s[k * TILE_D + seg * 4]), 0, 0);
      __builtin_amdgcn_global_load_async_to_lds_b128(
          (v4i*)(u + gRow + seg * 4), (v4i*)(&u_s[k * TILE_D + seg * 4]), 0, 0);
    }
    __builtin_amdgcn_s_wait_asynccnt(0);
  }
#else
  for (int i = tid; i < CHUNK_T * TILE_D; i += TPB) {
    const int k = i >> 6, dd = i & (TILE_D - 1);
    a_s[i] = a[rowStart + (size_t)k * D_ + dd];
    u_s[i] = u[rowStart + (size_t)k * D_ + dd];
  }
#endif
  __syncthreads();

  // ---- Per-thread local scan over 16 steps (zero initial state) ----
  const float x0v = (c == 0 && row == 0) ? x0[(size_t)b * D_ + d0 + dl] : 0.0f;
  float xl[TPT], pa[TPT];
  float p = 1.0f, x = 0.0f;
  const int tbase = row * TPT;
#pragma unroll
  for (int j = 0; j < TPT; ++j) {
    const float av = a_s[(tbase + j) * TILE_D + dl];
    float uv = u_s[(tbase + j) * TILE_D + dl];
    if (j == 0) uv = fmaf(av, x0v, uv);  // fold x0 into first step of the grid
    x = fmaf(av, x, uv);
    p *= av;
    xl[j] = x;
    pa[j] = p;
  }

  // ---- Combine the 4 rows through LDS: exclusive row-prefix per channel ----
  sP[row * TILE_D + dl] = p;
  sH[row * TILE_D + dl] = x;
  __syncthreads();
  float rp = 1.0f, rh = 0.0f;  // affine over rows [0, row)
  for (int r = 0; r < row; ++r) {
    const float Pr = sP[r * TILE_D + dl];
    const float Hr = sH[r * TILE_D + dl];
    rh = fmaf(Pr, rh, Hr);
    rp *= Pr;
  }

  // ---- Publish per-channel chunk aggregate (P_block, H_block), then flag ----
  const int node = (b * N_TILES + tile) * N_CHUNKS + c;
  if (row == ROWS - 1) {
    const float bp = rp * p;
    const float bh = fmaf(p, rh, x);
    __hip_atomic_store(&aggP[(size_t)node * TILE_D + dl], bp, __ATOMIC_RELAXED,
                       __HIP_MEMORY_SCOPE_AGENT);
    __hip_atomic_store(&aggH[(size_t)node * TILE_D + dl], bh, __ATOMIC_RELAXED,
                       __HIP_MEMORY_SCOPE_AGENT);
    __threadfence();
  }
  __syncthreads();
  if (tid == 0)
    __hip_atomic_store(&flags[node], 1, __ATOMIC_RELEASE, __HIP_MEMORY_SCOPE_AGENT);

  // ---- Deterministic lookback: fold ALL predecessor aggregates, fixed order ----
  if (c == 0) {
    if (tid < TILE_D) sHin[tid] = 0.0f;
  } else {
    float SP = 1.0f, SH = 0.0f;  // affine suffix over chunks (pch, c-1]
    const int nodeBase = (b * N_TILES + tile) * N_CHUNKS;
    for (int pch = c - 1; pch >= 0; --pch) {
      for (;;) {
        if (tid == 0)
          sFlag = __hip_atomic_load(&flags[nodeBase + pch], __ATOMIC_ACQUIRE,
                                    __HIP_MEMORY_SCOPE_AGENT);
        __syncthreads();
        const int f = sFlag;
        __syncthreads();
        if (f != 0) break;
        __builtin_amdgcn_s_sleep(2);
      }
      if (tid < TILE_D) {
        const float Pp = __hip_atomic_load(&aggP[(size_t)(nodeBase + pch) * TILE_D + tid],
                                           __ATOMIC_RELAXED, __HIP_MEMORY_SCOPE_AGENT);
        const float Hp = __hip_atomic_load(&aggH[(size_t)(nodeBase + pch) * TILE_D + tid],
                                           __ATOMIC_RELAXED, __HIP_MEMORY_SCOPE_AGENT);
        SH = fmaf(SP, Hp, SH);  // combine(agg_p, suffix)
        SP *= Pp;
      }
    }
    if (tid < TILE_D) sHin[tid] = SH;  // value entering this chunk (init state is 0)
  }
  __syncthreads();

  // ---- Apply carry and store ----
  const float hin = sHin[dl];
  const float sIn = fmaf(rp, hin, rh);  // value entering this thread's row
#pragma unroll
  for (int j = 0; j < TPT; ++j) {
    out[rowStart + (size_t)(tbase + j) * D_ + dl] = fmaf(pa[j], sIn, xl[j]);
  }
}

extern "C" void kernel_launch(void* const* d_in, const int* in_sizes, int n_in,
                              void* d_out, int out_size, void* d_ws, size_t ws_size,
                              hipStream_t stream) {
  const float* x0 = (const float*)d_in[0];  // [B, D]
  const float* a = (const float*)d_in[1];   // [B, T, D]
  const float* u = (const float*)d_in[2];   // [B, T, D]
  float* out = (float*)d_out;               // [B, T, D]

  const size_t nCh = (size_t)B_ * N_TILES * N_CHUNKS * TILE_D;  // 524288 aggregate slots
  const int nFlags = B_ * N_TILES * N_CHUNKS;                   // 8192
  const size_t need = 2 * nCh * sizeof(float) + (size_t)nFlags * sizeof(int);

  if (ws_size < need) {
    // Not enough scratch for lookback: correct (slower) sequential fallback.
    const int n = B_ * D_;
    hipLaunchKernelGGL(scan_seq_kernel, dim3((n + 255) / 256), dim3(256), 0, stream,
                       x0, a, u, out);
    return;
  }

  float* aggP = (float*)d_ws;
  float* aggH = aggP + nCh;
  int* flags = (int*)(aggH + nCh);

  hipLaunchKernelGGL(zero_flags_kernel, dim3((nFlags + 255) / 256), dim3(256), 0, stream,
                     flags, nFlags);
  dim3 grid(N_CHUNKS, N_TILES, B_);
  hipLaunchKernelGGL(scan_chunked_kernel, grid, dim3(TPB), 0, stream,
                     x0, a, u, out, aggP, aggH, flags);
}